// RelativeLearned2DPE_66151086293476
// MI455X (gfx1250) — compile-verified
//
#include <hip/hip_runtime.h>
#include <hip/hip_bf16.h>

typedef __attribute__((ext_vector_type(2))) float v2f;
typedef __attribute__((ext_vector_type(4))) float v4f;
typedef __attribute__((ext_vector_type(8))) float v8f;

#define DIM   64
#define NREL  127
#define NPAD  128
#define NROWS 32768   // B*H*W = 8*64*64

// ---------------------------------------------------------------------------
// Phase 1: Rel[row][m] = sum_k q[row][k] * emb[m][k]   (row-major outputs,
// N padded to 128). One wave computes a 16(M) x 128(N) strip using
// V_WMMA_F32_16X16X4_F32, 16 chained WMMAs per 16x16 tile (K=64).
// gridDim.x = 512, blockDim.x = 256 (8 waves) -> 4096 waves =
// 2048 M-tiles x 2 embeddings.
// ---------------------------------------------------------------------------
__global__ __launch_bounds__(256) void pe_gemm_wmma(
    const float* __restrict__ q,       // [32768][64]
    const float* __restrict__ emb_h,   // [127][64]
    const float* __restrict__ emb_w,   // [127][64]
    float* __restrict__ relh,          // [32768][128]
    float* __restrict__ relw)          // [32768][128]
{
    const int lane  = threadIdx.x & 31;
    const int wave  = threadIdx.x >> 5;
    const int gwave = blockIdx.x * 8 + wave;     // 0..4095
    const int which = gwave & 1;                 // 0 -> W embedding, 1 -> H
    const int mtile = gwave >> 1;                // 0..2047

    const float* __restrict__ emb = which ? emb_h : emb_w;
    float* __restrict__ out       = which ? relh : relw;

    const int half = lane >> 4;        // 0 or 1
    const int l16  = lane & 15;
    const int kb   = half * 2;         // A/B fragment K sub-offset (0 or 2)

    // ---- load A fragments once: 16 K-steps, float2 each -------------------
    const float* qrow = q + (size_t)(mtile * 16 + l16) * DIM;
    v2f a[16];
#pragma unroll
    for (int ks = 0; ks < 16; ++ks) {
        a[ks].x = qrow[ks * 4 + kb + 0];
        a[ks].y = qrow[ks * 4 + kb + 1];
    }

    const int mbase = mtile * 16 + half * 8;

#pragma unroll
    for (int nt = 0; nt < 8; ++nt) {
        const int n    = nt * 16 + l16;
        const int nn   = (n < NREL) ? n : 0;
        const float sc = (n < NREL) ? 1.0f : 0.0f;
        const float* erow = emb + (size_t)nn * DIM;

        // ---- B fragments for this N-tile ----------------------------------
        v2f b[16];
#pragma unroll
        for (int ks = 0; ks < 16; ++ks) {
            b[ks].x = erow[ks * 4 + kb + 0] * sc;
            b[ks].y = erow[ks * 4 + kb + 1] * sc;
        }

        // ---- 16 chained WMMAs: D = A_k * B_k + C --------------------------
        v8f c = {};
#pragma unroll
        for (int ks = 0; ks < 16; ++ks) {
            c = __builtin_amdgcn_wmma_f32_16x16x4_f32(
                    /*neg_a=*/false, a[ks],
                    /*neg_b=*/false, b[ks],
                    /*c_mod=*/(short)0, c,
                    /*reuse_a=*/false, /*reuse_b=*/false);
        }

        // ---- store D: VGPR i -> row (mbase+i), col n ----------------------
#pragma unroll
        for (int i = 0; i < 8; ++i) {
            out[(size_t)(mbase + i) * NPAD + n] = c[i];
        }
    }
}

// ---------------------------------------------------------------------------
// Phase 2: out[b,h,w,h',w'] = RelH[row][h'-h+63] + RelW[row][w'-w+63]
// One block per row=(b,h,w); 256 threads write the 64x64 tile (16 KiB) with
// nontemporal float4 stores. Shifted windows staged in LDS.
// gridDim.x = 32768, blockDim.x = 256.
// ---------------------------------------------------------------------------
__global__ __launch_bounds__(256) void pe_broadcast_add(
    const float* __restrict__ relh,    // [32768][128]
    const float* __restrict__ relw,    // [32768][128]
    float* __restrict__ out)           // [32768][4096]
{
    const int row = blockIdx.x;        // b*4096 + h*64 + w
    const int h   = (row >> 6) & 63;
    const int w   = row & 63;
    const int t   = threadIdx.x;

    __shared__ float sh[64];
    __shared__ float sw[64];

    if (t < 64) {
        sh[t] = relh[(size_t)row * NPAD + (t - h + 63)];        // m in [0,126]
    } else if (t < 128) {
        const int wp = t - 64;
        sw[wp] = relw[(size_t)row * NPAD + (wp - w + 63)];
    }
    __syncthreads();

    const int hp  = t >> 2;            // 0..63  (h')
    const int seg = t & 3;             // 0..3   (16 floats each)
    const float rh = sh[hp];

    v4f* op = (v4f*)(out + ((size_t)row << 12) + (size_t)hp * 64 + seg * 16);
#pragma unroll
    for (int j = 0; j < 4; ++j) {
        const int wp = seg * 16 + j * 4;
        v4f v;
        v.x = rh + sw[wp + 0];
        v.y = rh + sw[wp + 1];
        v.z = rh + sw[wp + 2];
        v.w = rh + sw[wp + 3];
        __builtin_nontemporal_store(v, op + j);
    }
}

// ---------------------------------------------------------------------------
extern "C" void kernel_launch(void* const* d_in, const int* in_sizes, int n_in,
                              void* d_out, int out_size, void* d_ws, size_t ws_size,
                              hipStream_t stream) {
    const float* q     = (const float*)d_in[0];   // [8,64,64,64]
    const float* emb_h = (const float*)d_in[1];   // [127,64]
    const float* emb_w = (const float*)d_in[2];   // [127,64]
    float* out = (float*)d_out;                   // [8,64,64,64,64]

    float* relh = (float*)d_ws;                   // 32768*128 floats = 16 MiB
    float* relw = relh + (size_t)NROWS * NPAD;    // next 16 MiB (needs 32 MiB ws)

    pe_gemm_wmma<<<512, 256, 0, stream>>>(q, emb_h, emb_w, relh, relw);
    pe_broadcast_add<<<NROWS, 256, 0, stream>>>(relh, relw, out);
}